// ChebConv_19172734009347
// MI455X (gfx1250) — compile-verified
//
#include <hip/hip_runtime.h>

// ---------------------------------------------------------------------------
// ChebConv (K=5) for MI455X / gfx1250, wave32.
//   V=196608 pixels, NNZ=9V COO Laplacian, B=4, FIN=32, FOUT=64.
//   Phase 1: x0 = transpose(inputs) -> [V, FIN*B] (rows L2-resident: 100MB)
//   Phase 2: Chebyshev recurrence via atomic-scatter SpMM (f32, in L2)
//   Phase 3: per-k rank-32 GEMM update into out using V_WMMA_F32_16X16X4_F32
// Workspace: 3 rotating x-buffers = 3 * V * 128 * 4 B = 302 MB.
// ---------------------------------------------------------------------------

#define Vn    196608
#define NNZn  (9 * Vn)
#define Bn    4
#define FINn  32
#define FOUTn 64
#define Kn    5
#define XC    (FINn * Bn)   /* 128 columns of each Chebyshev state */
#define EPB   4             /* edges per SpMM block */

typedef float v2f __attribute__((ext_vector_type(2)));
typedef float v8f __attribute__((ext_vector_type(8)));

// ---- inputs [B, V, FIN] -> x0 [V, FIN*B] with x0[v][fin*4 + b] --------------
__global__ __launch_bounds__(256) void k_transpose(const float* __restrict__ in,
                                                   float* __restrict__ x0) {
    int idx = blockIdx.x * 256 + threadIdx.x;      // read-coalesced over B*V*FIN
    int fin = idx & (FINn - 1);
    int v   = (idx / FINn) % Vn;
    int b   = idx / (Vn * FINn);
    x0[v * XC + fin * Bn + b] = in[idx];
}

// ---- y = beta * src   (src == nullptr -> y = 0) ----------------------------
__global__ __launch_bounds__(256) void k_scale(float* __restrict__ y,
                                               const float* __restrict__ src,
                                               float beta) {
    int idx = blockIdx.x * 256 + threadIdx.x;
    y[idx] = src ? beta * src[idx] : 0.0f;
}

// ---- y += alpha * L x  (COO gather + atomic scatter-add; x,y in L2) --------
__global__ __launch_bounds__(128) void k_spmm(const int*   __restrict__ rows,
                                              const int*   __restrict__ cols,
                                              const float* __restrict__ vals,
                                              const float* __restrict__ x,
                                              float*       __restrict__ y,
                                              float alpha) {
    const int tid = threadIdx.x;                   // 0..127 -> feature column
    const int e0  = blockIdx.x * EPB;
    if (tid == 0) {                                // pull next edge chunk into L2
        __builtin_prefetch(&rows[e0 + 64 * EPB], 0, 1);
        __builtin_prefetch(&cols[e0 + 64 * EPB], 0, 1);
        __builtin_prefetch(&vals[e0 + 64 * EPB], 0, 1);
    }
#pragma unroll
    for (int e = 0; e < EPB; ++e) {
        const int   edge = e0 + e;
        const int   r    = rows[edge];             // block-uniform -> scalar load
        const int   c    = cols[edge];
        const float w    = alpha * vals[edge];
        const float xv   = x[c * XC + tid];        // 512B contiguous gather
        unsafeAtomicAdd(&y[r * XC + tid], w * xv); // global_atomic_add_f32 @ L2
    }
}

// ---- out[bv, fout] += sum_fin Xk[v, fin*4+b] * W[fin*K + k][fout] ----------
// One block = 4 waves = one 16-row M tile; wave w covers N tile w (16 cols).
// WMMA f32 16x16x4: A = 16x4 (v2f), B = 4x16 (v2f), C/D = 16x16 (v8f).
__global__ __launch_bounds__(128) void k_cheb_gemm(const float* __restrict__ X,
                                                   const float* __restrict__ Wm,   // flat [K*FIN, FOUT]
                                                   const float* __restrict__ bias,
                                                   float*       __restrict__ out,  // [B*V, FOUT]
                                                   int k, int first) {
    __shared__ float Atile[16][FINn + 1];          // +1 pad: conflict-free column reads
    __shared__ float Wk[FINn][FOUTn];

    const int tid = threadIdx.x;
    const int bv0 = blockIdx.x * 16;               // tile never straddles b (V % 16 == 0)
    const int b   = bv0 / Vn;
    const int v0  = bv0 % Vn;

    // Stage this k's effective 32x64 weight slice: row fin <- Wm[fin*K + k]
    for (int i = tid; i < FINn * FOUTn; i += 128) {
        const int fin = i >> 6, c = i & 63;
        Wk[fin][c] = Wm[(fin * Kn + k) * FOUTn + c];
    }
    // Stage the 16x32 A tile: A[r][fin] = Xk[v0+r][fin*4 + b]
    for (int i = tid; i < 16 * FINn; i += 128) {
        const int r = i >> 5, fin = i & 31;
        Atile[r][fin] = X[(v0 + r) * XC + (fin << 2) + b];
    }
    __syncthreads();

    const int wave = tid >> 5;                     // N tile index 0..3
    const int lane = tid & 31;
    const int half = lane >> 4;                    // 0: lanes 0-15, 1: lanes 16-31
    const int l16  = lane & 15;
    const int n0   = wave * 16;

    // C/D layout: VGPR j, lane -> M = j + 8*half, N = l16
    v8f c;
    if (first) {
        const float bb = bias[n0 + l16];
#pragma unroll
        for (int j = 0; j < 8; ++j) c[j] = bb;
    } else {
#pragma unroll
        for (int j = 0; j < 8; ++j)
            c[j] = out[(size_t)(bv0 + j + half * 8) * FOUTn + n0 + l16];
    }

#pragma unroll
    for (int f0 = 0; f0 < FINn; f0 += 4) {
        // A 16x4 f32 layout: lane l16 holds M=l16; VGPR0/1 = K = f0+2*half (+1)
        v2f a, bf;
        a.x  = Atile[l16][f0 + 2 * half];
        a.y  = Atile[l16][f0 + 2 * half + 1];
        // B 4x16 f32 layout (mirrored roles): lane l16 holds N=l16; VGPR0/1 = K
        bf.x = Wk[f0 + 2 * half][n0 + l16];
        bf.y = Wk[f0 + 2 * half + 1][n0 + l16];
        c = __builtin_amdgcn_wmma_f32_16x16x4_f32(
                /*neg_a=*/false, a, /*neg_b=*/false, bf,
                /*c_mod=*/(short)0, c, /*reuse_a=*/false, /*reuse_b=*/false);
    }

#pragma unroll
    for (int j = 0; j < 8; ++j)
        out[(size_t)(bv0 + j + half * 8) * FOUTn + n0 + l16] = c[j];
}

// ---------------------------------------------------------------------------
extern "C" void kernel_launch(void* const* d_in, const int* in_sizes, int n_in,
                              void* d_out, int out_size, void* d_ws, size_t ws_size,
                              hipStream_t stream) {
    const int*   rows = (const int*)  d_in[0];
    const int*   cols = (const int*)  d_in[1];
    const float* vals = (const float*)d_in[2];
    const float* inp  = (const float*)d_in[3];
    const float* wgt  = (const float*)d_in[4];   // flat [K*FIN, FOUT]
    const float* bias = (const float*)d_in[5];
    float*       out  = (float*)d_out;

    const size_t xsz = (size_t)Vn * XC;          // floats per Chebyshev state
    float* xb0 = (float*)d_ws;                   // rotating x buffers (3 * 100.7 MB)
    float* xb1 = xb0 + xsz;
    float* xb2 = xb1 + xsz;

    const int elemGrid = (int)(xsz / 256);       // 98304
    const int spmmGrid = NNZn / EPB;             // 442368
    const int mTiles   = (Bn * Vn) / 16;         // 49152

    // x0 = transpose(inputs); out = bias + x0 @ W_{k=0}
    k_transpose<<<(Bn * Vn * FINn) / 256, 256, 0, stream>>>(inp, xb0);
    k_cheb_gemm<<<mTiles, 128, 0, stream>>>(xb0, wgt, bias, out, 0, 1);

    // x1 = L x0
    k_scale<<<elemGrid, 256, 0, stream>>>(xb1, nullptr, 0.0f);
    k_spmm<<<spmmGrid, 128, 0, stream>>>(rows, cols, vals, xb0, xb1, 1.0f);
    k_cheb_gemm<<<mTiles, 128, 0, stream>>>(xb1, wgt, bias, out, 1, 0);

    // x2 = 2 L x1 - x0
    k_scale<<<elemGrid, 256, 0, stream>>>(xb2, xb0, -1.0f);
    k_spmm<<<spmmGrid, 128, 0, stream>>>(rows, cols, vals, xb1, xb2, 2.0f);
    k_cheb_gemm<<<mTiles, 128, 0, stream>>>(xb2, wgt, bias, out, 2, 0);

    // x3 = 2 L x2 - x1   (reuse xb0: x0 fully consumed)
    k_scale<<<elemGrid, 256, 0, stream>>>(xb0, xb1, -1.0f);
    k_spmm<<<spmmGrid, 128, 0, stream>>>(rows, cols, vals, xb2, xb0, 2.0f);
    k_cheb_gemm<<<mTiles, 128, 0, stream>>>(xb0, wgt, bias, out, 3, 0);

    // x4 = 2 L x3 - x2   (reuse xb1)
    k_scale<<<elemGrid, 256, 0, stream>>>(xb1, xb2, -1.0f);
    k_spmm<<<spmmGrid, 128, 0, stream>>>(rows, cols, vals, xb0, xb1, 2.0f);
    k_cheb_gemm<<<mTiles, 128, 0, stream>>>(xb1, wgt, bias, out, 4, 0);
}